// CrossConv_65859028516953
// MI455X (gfx1250) — compile-verified
//
#include <hip/hip_runtime.h>
#include <cstdint>

// Depthwise per-sample 7x7 conv, SAME padding, NHWC fp32.
// inputs:  [32,128,128,128]  kernels: [32,7,7,128]  out: [32,128,128,128]

typedef float v2f __attribute__((ext_vector_type(2)));
typedef float v4f __attribute__((ext_vector_type(4)));

#define BATCH 32
#define HDIM  128
#define WDIM  128
#define CDIM  128
#define KHW   7
#define PADR  3

#define TH 16               // output rows per block
#define WT 32               // output cols per block
#define CT 16               // channels per block (8 float2 lanes)
#define INROWS (TH + 6)     // 22
#define INCOLS (WT + 6)     // 38
#define NCHUNK (INROWS * INCOLS * (CT / 4))   // 16-byte chunks in the LDS tile = 3344

__global__ __launch_bounds__(256)
void dwconv7x7_kernel(const float* __restrict__ in,
                      const float* __restrict__ kern,
                      float* __restrict__ out) {
    __shared__ float lds_in[INROWS * INCOLS * CT];   // 53,504 B

    const int tid = threadIdx.x;
    const int bx  = blockIdx.x;
    const int cb  = bx & 7;                // 8 channel tiles
    const int wb  = (bx >> 3) & 3;         // 4 w tiles
    const int hb  = (bx >> 5) & 7;         // 8 h tiles
    const int b   = bx >> 8;               // 32 batches
    const int c0  = cb * CT;
    const int w0  = wb * WT;
    const int h0  = hb * TH;

    const bool interior = (h0 >= PADR) && (h0 + TH + PADR <= HDIM) &&
                          (w0 >= PADR) && (w0 + WT + PADR <= WDIM);

    // ---- border tiles: pre-zero LDS so OOB pixels read as 0 ----
    if (!interior) {
        v4f z = {0.f, 0.f, 0.f, 0.f};
        for (int i = tid; i < NCHUNK; i += 256)
            ((v4f*)lds_in)[i] = z;
        __syncthreads();   // DS writes complete before async engine touches LDS
    }

    // ---- stage (22 x 38 x 16ch) halo tile into LDS with CDNA5 async copies ----
    for (int i = tid; i < NCHUNK; i += 256) {
        int q   = i & 3;            // 16B chunk inside the 64B channel slice
        int pix = i >> 2;
        int r   = pix / INCOLS;
        int col = pix - r * INCOLS;
        int gh  = h0 - PADR + r;
        int gw  = w0 - PADR + col;
        if ((unsigned)gh < (unsigned)HDIM && (unsigned)gw < (unsigned)WDIM) {
            const float* gptr = in +
                ((size_t)((b * HDIM + gh) * WDIM + gw) * CDIM + c0 + q * 4);
            uint32_t lds_off = (uint32_t)(uintptr_t)&lds_in[pix * CT + q * 4];
            asm volatile("global_load_async_to_lds_b128 %0, %1, off"
                         :: "v"(lds_off), "v"(gptr)
                         : "memory");
        }
    }
    asm volatile("s_wait_asynccnt 0" ::: "memory");
    __syncthreads();

    // ---- per-thread mapping: lane owns 2 consecutive channels, 1 output column ----
    const int cp = tid & 7;         // channel pair index (0..7)
    const int wi = tid >> 3;        // output column within tile (0..31)
    const v2f* lds2 = (const v2f*)lds_in;   // [pix][8] channel pairs

    // preload 49 weight pairs for this (b, channel-pair)
    v2f wt[KHW][KHW];
    const float* kbase = kern + (size_t)b * KHW * KHW * CDIM + c0 + 2 * cp;
#pragma unroll
    for (int kh = 0; kh < KHW; ++kh)
#pragma unroll
        for (int kw = 0; kw < KHW; ++kw)
            wt[kh][kw] = *(const v2f*)(kbase + (kh * KHW + kw) * CDIM);

    v2f acc[TH];
#pragma unroll
    for (int h = 0; h < TH; ++h) acc[h] = (v2f){0.f, 0.f};

    // stream input rows once; each row feeds up to 7 output rows (49 pk-FMAs / 7 DS loads)
#pragma unroll
    for (int r = 0; r < INROWS; ++r) {
        v2f in7[KHW];
#pragma unroll
        for (int j = 0; j < KHW; ++j)
            in7[j] = lds2[(r * INCOLS + wi + j) * (CT / 2) + cp];
#pragma unroll
        for (int kh = 0; kh < KHW; ++kh) {
            const int h = r - kh;                  // folds at compile time after unroll
            if (h >= 0 && h < TH) {
#pragma unroll
                for (int kw = 0; kw < KHW; ++kw)
                    acc[h] = __builtin_elementwise_fma(in7[kw], wt[kh][kw], acc[h]);
            }
        }
    }

    // ---- store 16 output rows (64B contiguous per (h,w) across channel pairs) ----
    float* obase = out + ((size_t)((b * HDIM + h0) * WDIM) + w0 + wi) * CDIM + c0 + 2 * cp;
#pragma unroll
    for (int h = 0; h < TH; ++h)
        *(v2f*)(obase + (size_t)h * WDIM * CDIM) = acc[h];
}

extern "C" void kernel_launch(void* const* d_in, const int* in_sizes, int n_in,
                              void* d_out, int out_size, void* d_ws, size_t ws_size,
                              hipStream_t stream) {
    const float* inputs  = (const float*)d_in[0];
    const float* kernels = (const float*)d_in[1];
    float*       outp    = (float*)d_out;

    // 32 batches * 8 h-tiles * 4 w-tiles * 8 c-tiles = 8192 blocks
    dim3 grid(BATCH * 8 * 4 * 8), block(256);
    hipLaunchKernelGGL(dwconv7x7_kernel, grid, block, 0, stream,
                       inputs, kernels, outp);
}